// LossCTC_11003706212600
// MI455X (gfx1250) — compile-verified
//
#include <hip/hip_runtime.h>
#include <hip/hip_bf16.h>

#define Bc 16
#define Tc 1000
#define Vc 4096
#define Sc 100
#define Lc 201      // 2*S+1
#define LP 224      // L padded to 7*32 for the single-wave scan
#define NEGF (-1.0e30f)

typedef float v2f __attribute__((ext_vector_type(2)));
typedef float v8f __attribute__((ext_vector_type(8)));
typedef int   v4i __attribute__((vector_size(16)));   // matches builtin param type

#if __has_builtin(__builtin_amdgcn_global_load_async_to_lds_b128) && __has_builtin(__builtin_amdgcn_s_wait_asynccnt)
#define CTC_ASYNC 1
#else
#define CTC_ASYNC 0
#endif

// -------------------------------------------------------------------------
// Pass 1: one block per (b,t) row. Stage 16KB row in LDS (async copy),
// compute logsumexp over V=4096 (max via shuffles, sum via WMMA f32 16x16x4
// K-collapse), then gather the 201 extended-label entries:
//   lp_z[b,t,s] = x[b,t,z[b,s]] - lse(b,t),  z[2i]=0 (blank), z[2i+1]=y[b,i]
// Pads s in [201,224) with NEG so the scan needs no bounds checks.
// -------------------------------------------------------------------------
__global__ __launch_bounds__(256) void ctc_row_kernel(const float* __restrict__ x,
                                                      const int* __restrict__ y,
                                                      float* __restrict__ lpz)
{
    __shared__ __align__(16) float srow[Vc];
    __shared__ float sred[8];
    __shared__ float sbc[2];

    const int tid  = threadIdx.x;
    const int lane = tid & 31;
    const int wid  = tid >> 5;
    const int row  = blockIdx.x;          // row = b*T + t
    const int b    = row / Tc;
    const float* __restrict__ xrow = x + (size_t)row * Vc;

    // ---- stage row into LDS: 256 threads x 4 x 16B = 16KB ----
#if CTC_ASYNC
    #pragma unroll
    for (int k = 0; k < 4; ++k) {
        const int off = k * 1024 + tid * 4;
        __builtin_amdgcn_global_load_async_to_lds_b128(
            (v4i*)(xrow + off),
            (__attribute__((address_space(3))) v4i*)(&srow[off]),
            0, 0);
    }
    __builtin_amdgcn_s_wait_asynccnt(0);
    __syncthreads();
#else
    #pragma unroll
    for (int k = 0; k < 4; ++k) {
        float4 v = ((const float4*)xrow)[tid + k * 256];
        ((float4*)srow)[tid + k * 256] = v;
    }
    __syncthreads();
#endif

    // ---- per-thread 16 values (bank-conflict-free strided reads) ----
    float xs[16];
    #pragma unroll
    for (int i = 0; i < 16; ++i) xs[i] = srow[tid + i * 256];

    // ---- row max: local -> wave shuffle tree -> cross-wave via LDS ----
    float wmax = xs[0];
    #pragma unroll
    for (int i = 1; i < 16; ++i) wmax = fmaxf(wmax, xs[i]);
    #pragma unroll
    for (int off = 16; off > 0; off >>= 1) wmax = fmaxf(wmax, __shfl_xor(wmax, off));
    if (lane == 0) sred[wid] = wmax;
    __syncthreads();
    if (tid == 0) {
        float m = sred[0];
        #pragma unroll
        for (int j = 1; j < 8; ++j) m = fmaxf(m, sred[j]);
        sbc[0] = m;
    }
    __syncthreads();
    const float rowmax = sbc[0];

    // ---- exp-sum: 2 partials/lane -> WMMA 16x16x4 collapses K=4 ----
    float p0 = 0.0f, p1 = 0.0f;
    #pragma unroll
    for (int i = 0; i < 8; ++i)  p0 += __expf(xs[i] - rowmax);
    #pragma unroll
    for (int i = 8; i < 16; ++i) p1 += __expf(xs[i] - rowmax);

    // A (16x4 f32): lane m<16 holds K0,K1; lane m+16 holds K2,K3 -> our
    // {p0,p1} pair per lane is exactly that striping. B = ones => D[m,*] =
    // sum of the 4 partials of "row" m. D VGPR r holds rowsum(r) on lanes
    // 0-15 and rowsum(r+8) on lanes 16-31, so summing the 8 components and
    // one xor-16 shuffle yields the full 64-value wave sum on every lane.
    v2f a;  a.x = p0;  a.y = p1;
    v2f bb; bb.x = 1.0f; bb.y = 1.0f;
    v8f c = {};
    v8f d = __builtin_amdgcn_wmma_f32_16x16x4_f32(false, a, false, bb,
                                                  (short)0, c, false, false);
    float wsum = d[0] + d[1] + d[2] + d[3] + d[4] + d[5] + d[6] + d[7];
    wsum += __shfl_xor(wsum, 16);

    if (lane == 0) sred[wid] = wsum;
    __syncthreads();
    if (tid == 0) {
        float s = sred[0];
        #pragma unroll
        for (int j = 1; j < 8; ++j) s += sred[j];
        sbc[1] = rowmax + __logf(s);
    }
    __syncthreads();
    const float lse = sbc[1];

    // ---- gather extended-label log-probs ----
    if (tid < LP) {
        const int s = tid;
        float val;
        if (s >= Lc) {
            val = NEGF;
        } else if (s & 1) {
            const int idx = y[b * Sc + ((s - 1) >> 1)];
            val = srow[idx] - lse;
        } else {
            val = srow[0] - lse;  // blank
        }
        lpz[(size_t)row * LP + s] = val;
    }
}

// -------------------------------------------------------------------------
// Pass 2: CTC forward recursion, one wave32 per batch element.
// alpha[s] lives in 7 VGPRs/lane (s = i*32 + lane). Shifts by 1/2 states
// are wave shuffles with chunk-boundary patch-up; no barriers in the loop.
// Loop runs t in [1, f_len) which is exactly the reference's freeze mask.
// -------------------------------------------------------------------------
__global__ __launch_bounds__(32) void ctc_scan_kernel(const float* __restrict__ lpz,
                                                      const int* __restrict__ y,
                                                      const int* __restrict__ f_len,
                                                      const int* __restrict__ y_len,
                                                      float* __restrict__ losses)
{
    __shared__ float sa[LP];
    const int b    = blockIdx.x;
    const int lane = threadIdx.x;
    const float* __restrict__ row0 = lpz + (size_t)b * Tc * LP;
    const int* __restrict__ yb = y + b * Sc;

    // allow_skip per state: s odd, and (s==1 or y[(s-1)/2] != y[(s-3)/2])
    bool allow[7];
    #pragma unroll
    for (int i = 0; i < 7; ++i) {
        const int s = i * 32 + lane;
        bool al = false;
        if ((s & 1) && s < Lc) {
            if (s == 1) al = true;
            else {
                const int j = (s - 1) >> 1;
                al = (yb[j] != yb[j - 1]);
            }
        }
        allow[i] = al;
    }

    // alpha init: alpha0[0]=lp[0,0], alpha0[1]=lp[0,1], rest NEG
    float alp[7];
    #pragma unroll
    for (int i = 0; i < 7; ++i) alp[i] = NEGF;
    if (lane < 2) alp[0] = row0[lane];

    const int fl = f_len[b];
    for (int t = 1; t < fl; ++t) {
        const float* __restrict__ rowt = row0 + (size_t)t * LP;
        if (t + 1 < fl) __builtin_prefetch(rowt + LP, 0, 0);

        float l[7];
        #pragma unroll
        for (int i = 0; i < 7; ++i) l[i] = rowt[i * 32 + lane];

        float na[7];
        #pragma unroll
        for (int i = 0; i < 7; ++i) {
            const float r31 = (i > 0) ? __shfl(alp[i - 1], 31) : NEGF;
            const float r30 = (i > 0) ? __shfl(alp[i - 1], 30) : NEGF;
            float s1 = __shfl_up(alp[i], 1);
            if (lane == 0) s1 = r31;
            float s2 = __shfl_up(alp[i], 2);
            if (lane == 0) s2 = r30;
            else if (lane == 1) s2 = r31;
            s2 = allow[i] ? s2 : NEGF;

            const float a1 = alp[i];
            const float m  = fmaxf(a1, fmaxf(s1, s2));
            const float lse = m + __logf(__expf(a1 - m) + __expf(s1 - m) + __expf(s2 - m));
            na[i] = lse + l[i];
        }
        #pragma unroll
        for (int i = 0; i < 7; ++i) alp[i] = na[i];
    }

    #pragma unroll
    for (int i = 0; i < 7; ++i) sa[i * 32 + lane] = alp[i];
    __syncthreads();

    if (lane == 0) {
        const int yl = y_len[b];
        const int i1 = 2 * yl;
        const int i2 = (2 * yl - 1 > 0) ? (2 * yl - 1) : 0;
        const float A1 = sa[i1], A2 = sa[i2];
        const float m  = fmaxf(A1, A2);
        float loss = -(m + __logf(__expf(A1 - m) + __expf(A2 - m)));
        if (!(loss < 1e29f)) loss = 0.0f;   // zero_infinity
        losses[b] = loss;
    }
}

// -------------------------------------------------------------------------
// Pass 3: mean over the 16 per-batch losses -> scalar output.
// -------------------------------------------------------------------------
__global__ void ctc_mean_kernel(const float* __restrict__ losses, float* __restrict__ out)
{
    if (threadIdx.x == 0) {
        float s = 0.0f;
        #pragma unroll
        for (int b = 0; b < Bc; ++b) s += losses[b];
        out[0] = s / (float)Bc;
    }
}

extern "C" void kernel_launch(void* const* d_in, const int* in_sizes, int n_in,
                              void* d_out, int out_size, void* d_ws, size_t ws_size,
                              hipStream_t stream) {
    (void)in_sizes; (void)n_in; (void)out_size; (void)ws_size;
    const float* x    = (const float*)d_in[0];   // [B,T,V] f32
    const int*   y    = (const int*)d_in[1];     // [B,S]   i32
    const int*   flen = (const int*)d_in[2];     // [B]     i32
    const int*   ylen = (const int*)d_in[3];     // [B]     i32
    float* out = (float*)d_out;

    float* wsf    = (float*)d_ws;
    float* losses = wsf;           // 16 floats
    float* lpz    = wsf + 64;      // [B,T,LP] f32 (~14.3 MB), 256B-aligned

    ctc_row_kernel <<<dim3(Bc * Tc), 256, 0, stream>>>(x, y, lpz);
    ctc_scan_kernel<<<dim3(Bc),       32, 0, stream>>>(lpz, y, flen, ylen, losses);
    ctc_mean_kernel<<<dim3(1),        32, 0, stream>>>(losses, out);
}